// TreeDecoder_20710332301745
// MI455X (gfx1250) — compile-verified
//
#include <hip/hip_runtime.h>

typedef __attribute__((ext_vector_type(2))) float v2f;
typedef __attribute__((ext_vector_type(8))) float v8f;

#define HDIM   1024
#define EDIM   512
#define NCLS   10000      // EOS == NCLS, vocab = NCLS+1
#define NNODE  256
#define LANN   512
#define NODE_PAD 272      // padded node rows so WMMA N-padding reads stay in-bounds

// ---------------------------------------------------------------------------
// Generic WMMA f32 GEMM:  C[M x N] (+)= A * B (+ bias[m]) (tanh)
//   A: ACOL ? A[k*lda+m] : A[m*lda+k]      (compile-time layout)
//   B: BCOL ? B[n*ldb+k] : B[k*ldb+n]
//   C: row-major C[m*ldc+n].  N multiple of 16, K multiple of 4, lda/ldb even.
// One wave computes one 16x16 C tile via V_WMMA_F32_16X16X4_F32.
// A-row index is clamped (not guarded): out-of-range rows only affect C rows
// that are never stored, so the hot loop is branch-free.
// ---------------------------------------------------------------------------
template <int ACOL, int BCOL>
__global__ __launch_bounds__(256) void gemm_wmma_f32(
    const float* __restrict__ A, int lda,
    const float* __restrict__ B, int ldb,
    float* __restrict__ C, int ldc,
    int M, int N, int K,
    const float* __restrict__ bias, int accum, int act)
{
  const int wave = threadIdx.x >> 5;
  const int lane = threadIdx.x & 31;
  const int tilesN = N >> 4;
  const int tilesM = (M + 15) >> 4;
  const int tile = blockIdx.x * (blockDim.x >> 5) + wave;
  if (tile >= tilesM * tilesN) return;          // wave-uniform exit
  const int tm = tile / tilesN;
  const int tn = tile % tilesN;
  const int half = lane >> 4;                   // 0/1
  const int l    = lane & 15;
  int mA = tm * 16 + l;                         // A fragment row (clamped)
  if (mA > M - 1) mA = M - 1;
  const int nB = tn * 16 + l;                   // B fragment col

  const float* __restrict__ Arow = ACOL ? (A + mA) : (A + (size_t)mA * lda);
  const float* __restrict__ Bcol = BCOL ? (B + (size_t)nB * ldb) : (B + nB);

  v8f acc = {0.f,0.f,0.f,0.f,0.f,0.f,0.f,0.f};
#pragma unroll 4
  for (int k0 = 0; k0 < K; k0 += 4) {
    const int ka = k0 + 2 * half;               // this lane-half's K pair (even)
    v2f a, b;
    if (ACOL) { a[0] = Arow[(size_t)ka * lda];
                a[1] = Arow[(size_t)(ka + 1) * lda]; }
    else      { a = *(const v2f*)(Arow + ka); }
    if (BCOL) { b = *(const v2f*)(Bcol + ka); }
    else      { b[0] = Bcol[(size_t)ka * ldb];
                b[1] = Bcol[(size_t)(ka + 1) * ldb]; }
    acc = __builtin_amdgcn_wmma_f32_16x16x4_f32(false, a, false, b,
                                                (short)0, acc, false, false);
  }
  // Epilogue: D VGPR r holds row (r + 8*half), col = lane&15
  for (int r = 0; r < 8; ++r) {
    const int m = tm * 16 + r + 8 * half;
    if (m >= M) break;
    const int n = tn * 16 + l;
    float v = acc[r];
    if (bias)  v += bias[m];
    if (accum) v += C[(size_t)m * ldc + n];
    if (act == 1) v = tanhf(v);
    C[(size_t)m * ldc + n] = v;
  }
}

// ---------------------------------------------------------------------------
__global__ void zero_f32(float* p, int n) {
  int i = blockIdx.x * blockDim.x + threadIdx.x;
  if (i < n) p[i] = 0.f;
}

__global__ void set_root(const float* __restrict__ rH, const float* __restrict__ rC,
                         float* Hbuf, float* Cbuf) {
  int m = blockIdx.x * blockDim.x + threadIdx.x;
  if (m < HDIM) { Hbuf[m] = rH[m]; Cbuf[m] = rC[m]; }
}

// X[(E+H) x XP] row-major; column j <-> parent pstart+j; padded cols zeroed.
__global__ void build_x(const float* __restrict__ emb, const int* __restrict__ values,
                        const float* __restrict__ etbuf,
                        float* __restrict__ X, int XP, int ncols, int pstart) {
  int j   = blockIdx.y;
  int row = blockIdx.x * blockDim.x + threadIdx.x;
  if (row >= EDIM + HDIM) return;
  float v = 0.f;
  if (j < ncols) {
    int p = pstart + j;
    v = (row < EDIM) ? emb[(size_t)values[p] * EDIM + row]
                     : etbuf[(size_t)p * HDIM + (row - EDIM)];
  }
  X[(size_t)row * XP + j] = v;
}

__device__ __forceinline__ float sigf(float x) { return 1.f / (1.f + expf(-x)); }

// gates G: 4 slot blocks of [4H x XP] row-major. Node t in level: slot=t&3, col=t>>2.
__global__ void lstm_cell(const float* __restrict__ G, int XP,
                          float* __restrict__ Hbuf, float* __restrict__ Cbuf,
                          int start, int end, int pstart) {
  int t = blockIdx.y;
  int i = start + t;
  if (i >= end) return;
  int s = t & 3, j = t >> 2;
  int m = blockIdx.x * blockDim.x + threadIdx.x;
  if (m >= HDIM) return;
  const float* Gs = G + (size_t)s * 4 * HDIM * XP;
  float ig = Gs[(size_t)(m)          * XP + j];
  float fg = Gs[(size_t)(m +   HDIM) * XP + j];
  float gg = Gs[(size_t)(m + 2*HDIM) * XP + j];
  float og = Gs[(size_t)(m + 3*HDIM) * XP + j];
  float cp = Cbuf[(size_t)(pstart + j) * HDIM + m];
  float c  = sigf(fg) * cp + sigf(ig) * tanhf(gg);
  float h  = sigf(og) * tanhf(c);
  Hbuf[(size_t)i * HDIM + m] = h;
  Cbuf[(size_t)i * HDIM + m] = c;
}

// softmax over L=512 rows of AL[512 x NP], one block per column (incl. pads).
__global__ void att_softmax(const float* __restrict__ AL, float* __restrict__ P, int NP) {
  int n = blockIdx.x, t = threadIdx.x;
  __shared__ float sh[256];
  float mx = -3.4e38f;
  for (int k = t; k < LANN; k += 256) mx = fmaxf(mx, AL[(size_t)k * NP + n]);
  sh[t] = mx; __syncthreads();
  for (int s = 128; s > 0; s >>= 1) { if (t < s) sh[t] = fmaxf(sh[t], sh[t+s]); __syncthreads(); }
  mx = sh[0]; __syncthreads();
  float sum = 0.f;
  for (int k = t; k < LANN; k += 256) sum += expf(AL[(size_t)k * NP + n] - mx);
  sh[t] = sum; __syncthreads();
  for (int s = 128; s > 0; s >>= 1) { if (t < s) sh[t] += sh[t+s]; __syncthreads(); }
  float inv = 1.f / sh[0];
  for (int k = t; k < LANN; k += 256)
    P[(size_t)k * NP + n] = expf(AL[(size_t)k * NP + n] - mx) * inv;
}

__global__ void scatter_et(const float* __restrict__ ET, int NP,
                           float* __restrict__ etbuf, int start, int nL) {
  int n = blockIdx.y; if (n >= nL) return;
  int m = blockIdx.x * blockDim.x + threadIdx.x; if (m >= HDIM) return;
  etbuf[(size_t)(start + n) * HDIM + m] = ET[(size_t)m * NP + n];
}

// Faithful double softmax: p = softmax(o); nl = logsumexp(p) - p[v]
__global__ void node_loss(const float* __restrict__ O, int NP,
                          const int* __restrict__ values,
                          int start, int nL, float* __restrict__ loss_arr) {
  int n = blockIdx.x; if (n >= nL) return;
  int t = threadIdx.x;
  __shared__ float sh[256];
  const int M = NCLS + 1;
  float mx = -3.4e38f;
  for (int k = t; k < M; k += 256) mx = fmaxf(mx, O[(size_t)k * NP + n]);
  sh[t] = mx; __syncthreads();
  for (int s = 128; s > 0; s >>= 1) { if (t < s) sh[t] = fmaxf(sh[t], sh[t+s]); __syncthreads(); }
  mx = sh[0]; __syncthreads();
  float s1 = 0.f;
  for (int k = t; k < M; k += 256) s1 += expf(O[(size_t)k * NP + n] - mx);
  sh[t] = s1; __syncthreads();
  for (int s = 128; s > 0; s >>= 1) { if (t < s) sh[t] += sh[t+s]; __syncthreads(); }
  float inv1 = 1.f / sh[0]; __syncthreads();
  float s2 = 0.f;                                 // sum of exp(prob_k), probs in [0,1]
  for (int k = t; k < M; k += 256) s2 += expf(expf(O[(size_t)k * NP + n] - mx) * inv1);
  sh[t] = s2; __syncthreads();
  for (int s = 128; s > 0; s >>= 1) { if (t < s) sh[t] += sh[t+s]; __syncthreads(); }
  if (t == 0) {
    int i = start + n;
    int v = values[i];
    float pv = expf(O[(size_t)v * NP + n] - mx) * inv1;
    float nl = logf(sh[0]) - pv;
    if (v == NCLS) nl *= 0.2f;
    loss_arr[i] = nl;
  }
}

__global__ void reduce_loss(const float* __restrict__ loss_arr, float* out) {
  __shared__ float sh[256];
  int t = threadIdx.x;
  sh[t] = loss_arr[t];
  __syncthreads();
  for (int s = 128; s > 0; s >>= 1) { if (t < s) sh[t] += sh[t+s]; __syncthreads(); }
  if (t == 0) out[0] = sh[0];
}

// ---------------------------------------------------------------------------
extern "C" void kernel_launch(void* const* d_in, const int* in_sizes, int n_in,
                              void* d_out, int out_size, void* d_ws, size_t ws_size,
                              hipStream_t stream) {
  (void)in_sizes; (void)n_in; (void)out_size; (void)ws_size;
  const float* rootH = (const float*)d_in[0];
  const float* rootC = (const float*)d_in[1];
  const float* ann   = (const float*)d_in[2];
  const int*   values= (const int*)  d_in[3];
  const float* emb   = (const float*)d_in[6];
  const float* W_ih  = (const float*)d_in[7];
  const float* W_hh  = (const float*)d_in[8];
  const float* b_ih  = (const float*)d_in[9];
  const float* b_hh  = (const float*)d_in[10];
  const float* W_att = (const float*)d_in[11];
  const float* b_att = (const float*)d_in[12];
  const float* W_pre = (const float*)d_in[13];
  const float* b_pre = (const float*)d_in[14];
  const float* W_out = (const float*)d_in[15];
  const float* b_out = (const float*)d_in[16];

  // workspace layout (floats)
  float* ws = (float*)d_ws;
  float* Hbuf  = ws;                              // NODE_PAD*H
  float* Cbuf  = Hbuf  + NODE_PAD * HDIM;
  float* etbuf = Cbuf  + NODE_PAD * HDIM;
  float* ahvT  = etbuf + NODE_PAD * HDIM;         // [H x L] row-major: ahvT[m][l]=ahv[l][m]
  float* Xm    = ahvT  + HDIM * LANN;             // [1536 x 48]
  float* Gm    = Xm    + (EDIM + HDIM) * 48;      // 4 slots x [4096 x 48]
  float* AL    = Gm    + 4 * 4 * HDIM * 48;       // [512 x 176]
  float* Pm    = AL    + LANN * 176;              // [512 x 176]
  float* CTX   = Pm    + LANN * 176;              // [1024 x 176]
  float* ET    = CTX   + HDIM * 176;              // [1024 x 176]
  float* Om    = ET    + HDIM * 176;              // [10001 x 176]
  float* lossA = Om    + (NCLS + 16) * 176;       // [256]

  auto gemm = [&](const float* A, int lda, int aCol,
                  const float* B, int ldb, int bCol,
                  float* C, int ldc, int M, int N, int K,
                  const float* bias, int accum, int act) {
    int tiles = ((M + 15) / 16) * (N / 16);
    dim3 g((tiles + 7) / 8), b(256);
    if (aCol) {
      if (bCol) gemm_wmma_f32<1,1><<<g,b,0,stream>>>(A,lda,B,ldb,C,ldc,M,N,K,bias,accum,act);
      else      gemm_wmma_f32<1,0><<<g,b,0,stream>>>(A,lda,B,ldb,C,ldc,M,N,K,bias,accum,act);
    } else {
      if (bCol) gemm_wmma_f32<0,1><<<g,b,0,stream>>>(A,lda,B,ldb,C,ldc,M,N,K,bias,accum,act);
      else      gemm_wmma_f32<0,0><<<g,b,0,stream>>>(A,lda,B,ldb,C,ldc,M,N,K,bias,accum,act);
    }
  };

  // shared per-level node stage: attention -> et -> output -> per-node loss
  auto node_stage = [&](int start, int nL) {
    int NP = ((nL + 15) / 16) * 16;
    // AL[l][n] = sum_m ahv[l][m] * H[start+n][m]   (A = ahvT col-major)
    gemm(ahvT, LANN, 1, Hbuf + (size_t)start * HDIM, HDIM, 1,
         AL, NP, LANN, NP, HDIM, nullptr, 0, 0);
    att_softmax<<<NP, 256, 0, stream>>>(AL, Pm, NP);
    // CTX[m][n] = sum_k ann[k][m] * P[k][n]        (A = annotations col-major)
    gemm(ann, HDIM, 1, Pm, NP, 0, CTX, NP, HDIM, NP, LANN, nullptr, 0, 0);
    // ET = tanh(W_pre[:, :H] @ h + W_pre[:, H:] @ ctx + b_pre)
    gemm(W_pre, 2 * HDIM, 0, Hbuf + (size_t)start * HDIM, HDIM, 1,
         ET, NP, HDIM, NP, HDIM, b_pre, 0, 0);
    gemm(W_pre + HDIM, 2 * HDIM, 0, CTX, NP, 0,
         ET, NP, HDIM, NP, HDIM, nullptr, 1, 1);
    scatter_et<<<dim3(HDIM / 256, nL), 256, 0, stream>>>(ET, NP, etbuf, start, nL);
    // O = W_out @ et + b_out   [10001 x NP]
    gemm(W_out, HDIM, 0, ET, NP, 0, Om, NP, NCLS + 1, NP, HDIM, b_out, 0, 0);
    node_loss<<<nL, 256, 0, stream>>>(Om, NP, values, start, nL, lossA);
  };

  // 1) zero node buffers (padding rows must be 0 for padded WMMA columns)
  {
    int n = 3 * NODE_PAD * HDIM;
    zero_f32<<<(n + 255) / 256, 256, 0, stream>>>(Hbuf, n);
  }
  // 2) precompute ahvT[m][l] = sum_k W_att[m][k]*ann[l][k] + b_att[m]
  gemm(W_att, HDIM, 0, ann, HDIM, 1, ahvT, LANN, HDIM, LANN, HDIM, b_att, 0, 0);
  // 3) level 0: root state, then attention/output for node 0
  set_root<<<HDIM / 256, 256, 0, stream>>>(rootH, rootC, Hbuf, Cbuf);
  node_stage(0, 1);
  // 4) levels 1..4.  For level [start,end): pstart=(start-1)/4, node start+4j+s
  //    has slot s and parent pstart+j (slot-independent => shared X matrix).
  struct Lvl { int start, end, pstart, xcols, XP; };
  const Lvl LV[4] = { {1, 5, 0, 1, 16}, {5, 21, 1, 4, 16},
                      {21, 85, 5, 16, 16}, {85, 256, 21, 43, 48} };
  for (int li = 0; li < 4; ++li) {
    const Lvl& L = LV[li];
    const int nL = L.end - L.start;
    // shared input matrix X = [emb(values[p]); et[p]]  (1536 x XP)
    build_x<<<dim3((EDIM + HDIM + 255) / 256, L.XP), 256, 0, stream>>>(
        emb, values, etbuf, Xm, L.XP, L.xcols, L.pstart);
    // gates per slot: G_s = W_ih[s]@X + b_ih[s]  then += W_hh[s]@Hparents + b_hh[s]
    for (int s = 0; s < 4; ++s) {
      float* Gs = Gm + (size_t)s * 4 * HDIM * L.XP;
      gemm(W_ih + (size_t)s * 4 * HDIM * (EDIM + HDIM), EDIM + HDIM, 0,
           Xm, L.XP, 0, Gs, L.XP, 4 * HDIM, L.XP, EDIM + HDIM,
           b_ih + (size_t)s * 4 * HDIM, 0, 0);
      gemm(W_hh + (size_t)s * 4 * HDIM * HDIM, HDIM, 0,
           Hbuf + (size_t)L.pstart * HDIM, HDIM, 1, Gs, L.XP, 4 * HDIM, L.XP, HDIM,
           b_hh + (size_t)s * 4 * HDIM, 1, 0);
    }
    lstm_cell<<<dim3(HDIM / 256, nL), 256, 0, stream>>>(
        Gm, L.XP, Hbuf, Cbuf, L.start, L.end, L.pstart);
    node_stage(L.start, nL);
  }
  // 5) deterministic tree-sum of the 256 per-node losses
  reduce_loss<<<1, 256, 0, stream>>>(lossA, (float*)d_out);
}